// _ConvMoE_1005022347655
// MI455X (gfx1250) — compile-verified
//
#include <hip/hip_runtime.h>
#include <hip/hip_bf16.h>
#include <math.h>

typedef __attribute__((ext_vector_type(2))) float v2f;
typedef __attribute__((ext_vector_type(8))) float v8f;

#define B_  16
#define H_  32
#define W_  32
#define C_  128
#define F_  128
#define E_  8
#define PH_ 16
#define PW_ 16
#define G_  (F_ * PH_ * PW_)      // 32768 gating flatten size
#define KPB_ (9 * C_ * F_)        // 147456 floats per mixed per-batch kernel
#define TAPSZ_ (32 * F_ * 4)      // floats per tap in swizzled weight layout

// Swizzled weight layout (WMMA-B native): w[tap][c/4][f][c%4]
// -> the two K values lane needs (c0+2*half, +1) are adjacent floats: b64 load.

// ---------------------------------------------------------------------------
// 3x3 SAME conv as implicit GEMM on V_WMMA_F32_16X16X4_F32.
// grid = B*H*(W/16) blocks, 256 threads (8 wave32). Block owns 16 output
// pixels of one (b,h); wave w owns f-tile [16w, 16w+16). K = 9*128 = 1152.
// ---------------------------------------------------------------------------
__global__ __launch_bounds__(256)
void conv3x3_wmma_f32(const float* __restrict__ x,
                      const float* __restrict__ wk, long wk_bstride,
                      const float* __restrict__ bias, int bias_bstride,
                      float* __restrict__ out)
{
  __shared__ float xs[3 * 18 * C_];          // 27648 B halo strip, zero-padded

  const int blk = blockIdx.x;                // b*64 + h*2 + wt
  const int wt  = blk & 1;
  const int h   = (blk >> 1) & (H_ - 1);
  const int b   = blk >> 6;
  const int w0  = wt * 16;
  const int tid = threadIdx.x;

  // cooperative halo load: rows h-1..h+1, cols w0-1..w0+16, all C
  for (int i = tid; i < 3 * 18 * C_; i += 256) {
    const int c   = i & (C_ - 1);
    const int col = (i >> 7) % 18;
    const int row = (i >> 7) / 18;
    const int y   = h + row - 1;
    const int xw  = w0 + col - 1;
    float v = 0.0f;
    if ((unsigned)y < (unsigned)H_ && (unsigned)xw < (unsigned)W_)
      v = x[(((size_t)b * H_ + y) * W_ + xw) * C_ + c];
    xs[i] = v;
  }
  __syncthreads();

  const int lane  = tid & 31;                // wave32
  const int wave  = tid >> 5;                // 0..7 -> f tile
  const int m     = lane & 15;               // A row / B col
  const int half  = lane >> 4;               // K half select
  const int fbase = wave * 16;
  const float* wb = wk + (size_t)b * wk_bstride;

  v8f acc0 = {}, acc1 = {};                  // two chains: break WMMA RAW dep
  for (int kh = 0; kh < 3; ++kh) {
    for (int kw = 0; kw < 3; ++kw) {
      const int tap = kh * 3 + kw;
      const float* ap = &xs[((kh * 18) + m + kw) * C_];               // A: LDS
      const float* bp = wb + (size_t)tap * TAPSZ_ + (fbase + m) * 4 + 2 * half;
      __builtin_prefetch(bp + TAPSZ_, 0, 1); // next tap -> global_prefetch_b8
      #pragma unroll 2
      for (int c0 = 0; c0 < C_; c0 += 8) {
        const int cA0 = c0 + 2 * half;
        const int cA1 = cA0 + 4;
        v2f a0, a1;
        a0[0] = ap[cA0];     a0[1] = ap[cA0 + 1];
        a1[0] = ap[cA1];     a1[1] = ap[cA1 + 1];
        const v2f b0 = *reinterpret_cast<const v2f*>(bp + (size_t)(c0 >> 2) * (F_ * 4));
        const v2f b1 = *reinterpret_cast<const v2f*>(bp + (size_t)((c0 >> 2) + 1) * (F_ * 4));
        acc0 = __builtin_amdgcn_wmma_f32_16x16x4_f32(
            false, a0, false, b0, (short)0, acc0, false, false);
        acc1 = __builtin_amdgcn_wmma_f32_16x16x4_f32(
            false, a1, false, b1, (short)0, acc1, false, false);
      }
    }
  }

  // D layout: vgpr r -> pixel m_out = r + 8*half, channel f = fbase + (lane&15)
  const int f = fbase + m;
  const float bv = bias[(size_t)b * bias_bstride + f];
  #pragma unroll
  for (int r = 0; r < 8; ++r) {
    const int mo = r + 8 * half;
    out[(((size_t)b * H_ + h) * W_ + (w0 + mo)) * F_ + f] = acc0[r] + acc1[r] + bv;
  }
}

// ---------------------------------------------------------------------------
// Repack [3,3,C,F] weights into WMMA-B swizzled layout [tap][c/4][f][c%4]
// ---------------------------------------------------------------------------
__global__ void repack_weights_k(const float* __restrict__ wsrc,
                                 float* __restrict__ wdst)
{
  const int i = blockIdx.x * blockDim.x + threadIdx.x;   // source order: f fastest
  if (i >= KPB_) return;
  const int f   = i & (F_ - 1);
  const int c   = (i >> 7) & (C_ - 1);
  const int tap = i >> 14;
  wdst[((size_t)(tap * 32 + (c >> 2)) * F_ + f) * 4 + (c & 3)] = wsrc[i];
}

// ---------------------------------------------------------------------------
// 2x2 maxpool stride 2 on [B,H,W,F] -> [B,16,16,F]
// ---------------------------------------------------------------------------
__global__ void maxpool2x2_k(const float* __restrict__ g, float* __restrict__ gp)
{
  const int i = blockIdx.x * blockDim.x + threadIdx.x;
  if (i >= B_ * PH_ * PW_ * F_) return;
  const int f  = i & (F_ - 1);
  const int pw = (i >> 7) & (PW_ - 1);
  const int ph = (i >> 11) & (PH_ - 1);
  const int b  = i >> 15;
  const float* p = g + (((size_t)b * H_ + 2 * ph) * W_ + 2 * pw) * F_ + f;
  const float v0 = p[0], v1 = p[F_], v2 = p[W_ * F_], v3 = p[W_ * F_ + F_];
  gp[i] = fmaxf(fmaxf(v0, v1), fmaxf(v2, v3));
}

// ---------------------------------------------------------------------------
// logits[b,e] = gp_flat[b,:] . Wo[:,e] + bo[e] + 1e-4*eps[b,e]; one block/(b,e)
// ---------------------------------------------------------------------------
__global__ __launch_bounds__(256)
void gate_logits_k(const float* __restrict__ gp, const float* __restrict__ wo,
                   const float* __restrict__ bo, const float* __restrict__ eps,
                   float* __restrict__ logits)
{
  __shared__ float red[256];
  const int be = blockIdx.x;                 // b*8+e
  const int e = be & (E_ - 1), b = be >> 3;
  const int t = threadIdx.x;
  const float* gpb = gp + (size_t)b * G_;
  float s = 0.0f;
  for (int i = t; i < G_; i += 256)
    s += gpb[i] * wo[(size_t)i * E_ + e];
  red[t] = s;
  __syncthreads();
  for (int off = 128; off > 0; off >>= 1) {
    if (t < off) red[t] += red[t + off];
    __syncthreads();
  }
  if (t == 0) logits[be] = red[0] + bo[e] + 1e-4f * eps[be];
}

// ---------------------------------------------------------------------------
// top-2 softmax gating: one thread per batch row (earliest index wins ties,
// matching lax.top_k)
// ---------------------------------------------------------------------------
__global__ void topk_gates_k(const float* __restrict__ logits,
                             float* __restrict__ gates)
{
  const int b = threadIdx.x;
  if (b >= B_) return;
  float v[E_];
  #pragma unroll
  for (int e = 0; e < E_; ++e) v[e] = logits[b * E_ + e];
  int i0 = 0; float m0 = v[0];
  #pragma unroll
  for (int e = 1; e < E_; ++e) if (v[e] > m0) { m0 = v[e]; i0 = e; }
  int i1 = -1; float m1 = -INFINITY;
  #pragma unroll
  for (int e = 0; e < E_; ++e) if (e != i0 && v[e] > m1) { m1 = v[e]; i1 = e; }
  const float e1 = expf(m1 - m0);
  const float inv = 1.0f / (1.0f + e1);
  #pragma unroll
  for (int e = 0; e < E_; ++e) gates[b * E_ + e] = 0.0f;
  gates[b * E_ + i0] = inv;
  gates[b * E_ + i1] = e1 * inv;
}

// ---------------------------------------------------------------------------
// cb[b,f] = sum_e gates[b,e] * expert_bias[f,e]
// ---------------------------------------------------------------------------
__global__ void combine_bias_k(const float* __restrict__ gates,
                               const float* __restrict__ ebias,
                               float* __restrict__ cb)
{
  const int i = blockIdx.x * blockDim.x + threadIdx.x;
  if (i >= B_ * F_) return;
  const int f = i & (F_ - 1), b = i >> 7;
  float s = 0.0f;
  #pragma unroll
  for (int e = 0; e < E_; ++e) s += gates[b * E_ + e] * ebias[f * E_ + e];
  cb[i] = s;
}

// ---------------------------------------------------------------------------
// kb[b, swizzle(kh,kw,c,f)] = sum_e gates[b,e]*(mu + softplus(rho)*eps)[...,e]
// rho has no kh dim (shape [3,C,F,E], broadcast over kh). mu/eps read once.
// Writes directly in the WMMA-B swizzled layout.
// ---------------------------------------------------------------------------
__global__ __launch_bounds__(256)
void combine_kernels_k(const float* __restrict__ gates,
                       const float* __restrict__ mu,
                       const float* __restrict__ rho,
                       const float* __restrict__ epsk,
                       float* __restrict__ kb)
{
  const int idx = blockIdx.x * blockDim.x + threadIdx.x;   // (kh,kw,c,f), f fastest
  if (idx >= KPB_) return;
  const int ridx = idx % (3 * C_ * F_);                    // (kw,c,f)
  const int f    = idx & (F_ - 1);
  const int c    = (idx >> 7) & (C_ - 1);
  const int tap  = idx >> 14;
  const size_t didx = ((size_t)(tap * 32 + (c >> 2)) * F_ + f) * 4 + (c & 3);

  const float4* mu4 = reinterpret_cast<const float4*>(mu + (size_t)idx * E_);
  const float4* ep4 = reinterpret_cast<const float4*>(epsk + (size_t)idx * E_);
  const float4* rh4 = reinterpret_cast<const float4*>(rho + (size_t)ridx * E_);
  float4 muv[2] = {mu4[0], mu4[1]};
  float4 epv[2] = {ep4[0], ep4[1]};
  float4 rhv[2] = {rh4[0], rh4[1]};
  const float* muf = reinterpret_cast<const float*>(muv);
  const float* epf = reinterpret_cast<const float*>(epv);
  const float* rhf = reinterpret_cast<const float*>(rhv);

  float kv[E_];
  #pragma unroll
  for (int e = 0; e < E_; ++e) {
    const float r  = rhf[e];
    const float sp = (r > 20.0f) ? r : log1pf(expf(r));    // softplus
    kv[e] = muf[e] + sp * epf[e];
  }
  #pragma unroll
  for (int b = 0; b < B_; ++b) {
    float s = 0.0f;
    #pragma unroll
    for (int e = 0; e < E_; ++e) s += gates[b * E_ + e] * kv[e];
    kb[(size_t)b * KPB_ + didx] = s;
  }
}

// ---------------------------------------------------------------------------
extern "C" void kernel_launch(void* const* d_in, const int* in_sizes, int n_in,
                              void* d_out, int out_size, void* d_ws, size_t ws_size,
                              hipStream_t stream)
{
  (void)in_sizes; (void)n_in; (void)out_size; (void)ws_size;
  const float* x    = (const float*)d_in[0];   // [16,32,32,128]
  const float* epsk = (const float*)d_in[1];   // [3,3,128,128,8]
  const float* epsg = (const float*)d_in[2];   // [16,8]
  const float* mu   = (const float*)d_in[3];   // [3,3,128,128,8]
  const float* rho  = (const float*)d_in[4];   // [3,128,128,8]
  const float* eb   = (const float*)d_in[5];   // [128,8]
  const float* gk   = (const float*)d_in[6];   // [3,3,128,128]
  const float* gb   = (const float*)d_in[7];   // [128]
  const float* gok  = (const float*)d_in[8];   // [32768,8]
  const float* gob  = (const float*)d_in[9];   // [8]
  float* out = (float*)d_out;                  // [16,32,32,128]

  float* ws     = (float*)d_ws;
  float* g      = ws;                                   // 2,097,152 f
  float* gp     = g + (size_t)B_ * H_ * W_ * F_;        //   524,288 f
  float* logits = gp + (size_t)B_ * PH_ * PW_ * F_;     //       128 f
  float* gates  = logits + B_ * E_;                     //       128 f
  float* cb     = gates + B_ * E_;                      //     2,048 f
  float* gkt    = cb + B_ * F_;                         //   147,456 f (swizzled gating w)
  float* kb     = gkt + KPB_;                           // 2,359,296 f (~21 MB total)

  const int conv_blocks = B_ * H_ * (W_ / 16);          // 1024

  // gating path
  repack_weights_k<<<(KPB_ + 255) / 256, 256, 0, stream>>>(gk, gkt);
  conv3x3_wmma_f32<<<conv_blocks, 256, 0, stream>>>(x, gkt, 0, gb, 0, g);
  maxpool2x2_k<<<(B_ * PH_ * PW_ * F_ + 255) / 256, 256, 0, stream>>>(g, gp);
  gate_logits_k<<<B_ * E_, 256, 0, stream>>>(gp, gok, gob, epsg, logits);
  topk_gates_k<<<1, 32, 0, stream>>>(logits, gates);

  // expert path: mix kernels/bias with gates, then one per-batch conv
  combine_bias_k<<<(B_ * F_ + 255) / 256, 256, 0, stream>>>(gates, eb, cb);
  combine_kernels_k<<<(KPB_ + 255) / 256, 256, 0, stream>>>(gates, mu, rho, epsk, kb);
  conv3x3_wmma_f32<<<conv_blocks, 256, 0, stream>>>(x, kb, KPB_, cb, F_, out);
}